// BOTNet_43576738185434
// MI455X (gfx1250) — compile-verified
//
#include <hip/hip_runtime.h>
#include <math.h>

#define Nn   10000
#define Ee   160000
#define Kc   128
#define NE_T 10
#define NBb  8
#define Gg   16
#define RMAXf 5.0f
#define AVGf 16.0f
#define MLPDc 16
#define SQ3f 1.7320508075688772f
#define PI_F 3.14159265358979323846f

typedef float v2f __attribute__((ext_vector_type(2)));
typedef float v8f __attribute__((ext_vector_type(8)));

// ---------------------------------------------------------------------------
// fp32 WMMA GEMM, fully unrolled, A-tile staged in LDS, immediate-offset B.
//   OUT[row, col] (=|+=) alpha * sum_k A[row,k] * B(k,col),  OUT is N x Kc.
//   B(k,c) = TRANSW ? W[c*LDW + k] : W[k*LDW + c]
// Requirements (all satisfied by our call sites):
//   - nrows multiple of 16 (Nn = 625*16), gridDim.x = nrows/16
//   - A row-major with lda == KD (tile is one contiguous 16*KD block)
//   - 256 threads = 8 waves covering the 8 column tiles of Kc=128
// ---------------------------------------------------------------------------
template<int KD, int LDW, int TRANSW>
__global__ void gemm_wmma_t(const float* __restrict__ A,
                            const float* __restrict__ W,
                            float* __restrict__ OUT,
                            float alpha, int accumulate)
{
    __shared__ float sA[16 * KD];
    const int tid  = threadIdx.x;
    const int lane = tid & 31;
    const int wave = tid >> 5;
    const int row0 = blockIdx.x * 16;

    // cooperative coalesced copy of the contiguous 16xKD A tile into LDS
    {
        const float4* src = (const float4*)(A + (size_t)row0 * KD);
        float4* dst = (float4*)sA;
        const int nvec = (16 * KD) / 4;
        for (int i = tid; i < nvec; i += 256) dst[i] = src[i];
    }
    __syncthreads();

    const int col0 = wave * 16;
    const int m    = lane & 15;
    const int kb   = (lane >> 4) << 1;      // K sub-offset 0 or 2
    const int ccol = col0 + m;
    // single base pointer per lane; all loop accesses use constant offsets
    const float* __restrict__ Wb = TRANSW ? (W + ccol * LDW) : (W + ccol);
    const float* __restrict__ Ab = sA + m * KD + kb;
    v8f acc = {0.f,0.f,0.f,0.f,0.f,0.f,0.f,0.f};

#pragma unroll
    for (int kk = 0; kk < KD; kk += 4) {
        v2f a = *(const v2f*)(Ab + kk);
        v2f b;
        if (TRANSW) {
            b = *(const v2f*)(Wb + kk + kb);
        } else {
            b.x = Wb[(kk + kb)     * LDW];
            b.y = Wb[(kk + kb + 1) * LDW];
        }
        acc = __builtin_amdgcn_wmma_f32_16x16x4_f32(false, a, false, b,
                                                    (short)0, acc, false, false);
    }
    const int rhalf = (lane >> 4) * 8;
    float* __restrict__ Ob = OUT + (row0 + rhalf) * Kc + ccol;
    if (accumulate) {
#pragma unroll
        for (int g = 0; g < 8; ++g) Ob[g * Kc] += acc[g] * alpha;
    } else {
#pragma unroll
        for (int g = 0; g < 8; ++g) Ob[g * Kc]  = acc[g] * alpha;
    }
}

// ---------------------------------------------------------------------------
// z-indexed skip GEMM:  OUT[n] += A[n] @ (TRANSW ? Wsk[z[n]]^T : Wsk[z[n]])
// NE_T masked WMMA passes; A tile staged once in LDS and reused by all passes.
// ---------------------------------------------------------------------------
template<int TRANSW>
__global__ void gemm_sk_t(const float* __restrict__ A,
                          const float* __restrict__ Wsk,   // NE_T * Kc * Kc
                          const int*   __restrict__ z,
                          float* __restrict__ OUT)
{
    __shared__ float sA[16 * Kc];
    __shared__ int   sZ[16];
    const int tid  = threadIdx.x;
    const int lane = tid & 31;
    const int wave = tid >> 5;
    const int row0 = blockIdx.x * 16;
    {
        const float4* src = (const float4*)(A + (size_t)row0 * Kc);
        float4* dst = (float4*)sA;
        for (int i = tid; i < (16 * Kc) / 4; i += 256) dst[i] = src[i];
    }
    if (tid < 16) sZ[tid] = z[row0 + tid];
    __syncthreads();

    const int col0 = wave * 16;
    const int m    = lane & 15;
    const int kb   = (lane >> 4) << 1;
    const int ccol = col0 + m;
    const int zr   = sZ[m];
    const float* __restrict__ Ab = sA + m * Kc + kb;
    v8f acc = {0.f,0.f,0.f,0.f,0.f,0.f,0.f,0.f};

    for (int i = 0; i < NE_T; ++i) {
        const float* __restrict__ Wb = Wsk + (size_t)i * Kc * Kc
                                     + (TRANSW ? ccol * Kc : ccol);
        const float msk = (zr == i) ? 1.f : 0.f;
#pragma unroll
        for (int kk = 0; kk < Kc; kk += 4) {
            v2f a = *(const v2f*)(Ab + kk);
            a.x *= msk; a.y *= msk;
            v2f b;
            if (TRANSW) {
                b = *(const v2f*)(Wb + kk + kb);
            } else {
                b.x = Wb[(kk + kb)     * Kc];
                b.y = Wb[(kk + kb + 1) * Kc];
            }
            acc = __builtin_amdgcn_wmma_f32_16x16x4_f32(false, a, false, b,
                                                        (short)0, acc, false, false);
        }
    }
    const int rhalf = (lane >> 4) * 8;
    float* __restrict__ Ob = OUT + (row0 + rhalf) * Kc + ccol;
#pragma unroll
    for (int g = 0; g < 8; ++g) Ob[g * Kc] += acc[g];
}

// ---------------------------------------------------------------------------
// small helpers
// ---------------------------------------------------------------------------
__device__ __forceinline__ float wave_red(float v) {
#pragma unroll
    for (int mm = 16; mm >= 1; mm >>= 1) v += __shfl_xor(v, mm, 32);
    return v;
}

// d(ef_b)/dr  (Bessel * polynomial cutoff, P=5)
__device__ __forceinline__ void ef_deriv(float r, float* def) {
    const float Cb = 0.6324555320336759f;          // sqrt(2/R_MAX)
    float x = r / RMAXf;
    float f, dfr;
    if (x < 1.0f) {
        float x2 = x*x, x4 = x2*x2, x5 = x4*x, x6 = x5*x, x7 = x6*x;
        f   = 1.f - 21.f*x5 + 35.f*x6 - 15.f*x7;
        dfr = (-105.f*x4 + 210.f*x5 - 105.f*x6) * (1.f / RMAXf);
    } else { f = 0.f; dfr = 0.f; }
    float invr = 1.f / r;
#pragma unroll
    for (int b = 0; b < NBb; ++b) {
        float a = (float)(b + 1) * PI_F / RMAXf;
        float s = __sinf(a * r), c = __cosf(a * r);
        float bess  = Cb * s * invr;
        float dbess = Cb * (a * c * invr - s * invr * invr);
        def[b] = dbess * f + bess * dfr;
    }
}

__global__ void zero_k(float* p, size_t n) {
    size_t i = (size_t)blockIdx.x * blockDim.x + threadIdx.x;
    if (i < n) p[i] = 0.f;
}

__global__ void embed_k(const int* __restrict__ z, const float* __restrict__ W_embed,
                        float* __restrict__ s) {
    int i = blockIdx.x * blockDim.x + threadIdx.x;
    if (i >= Nn * Kc) return;
    int n = i >> 7, k = i & 127;
    s[i] = W_embed[z[n] * Kc + k];
}

// per-edge geometry: r and radial features ef[8]
__global__ void edge_geom_k(const int* __restrict__ ei, const float* __restrict__ pos,
                            const float* __restrict__ shifts,
                            float* __restrict__ r_out, float* __restrict__ ef_out)
{
    int e = blockIdx.x * blockDim.x + threadIdx.x;
    if (e >= Ee) return;
    int snd = ei[e], rcv = ei[Ee + e];
    float vx = pos[rcv*3+0] - pos[snd*3+0] + shifts[e*3+0];
    float vy = pos[rcv*3+1] - pos[snd*3+1] + shifts[e*3+1];
    float vz = pos[rcv*3+2] - pos[snd*3+2] + shifts[e*3+2];
    float r = sqrtf(vx*vx + vy*vy + vz*vz + 1e-12f);
    r_out[e] = r;
    float x = r / RMAXf;
    float f;
    if (x < 1.0f) {
        float x5 = x*x*x*x*x;
        f = 1.f - 21.f*x5 + 35.f*x5*x - 15.f*x5*x*x;
    } else f = 0.f;
    const float Cb = 0.6324555320336759f;
    float invr = 1.f / r;
#pragma unroll
    for (int b = 0; b < NBb; ++b)
        ef_out[(size_t)e * NBb + b] = Cb * __sinf((float)(b+1) * PI_F * r / RMAXf) * invr * f;
}

// layer-1 forward edge messages: scatter m0 (K) and m1 (3*K) into node sums
__global__ void edge_fwd1_k(const int* __restrict__ ei, const float* __restrict__ pos,
                            const float* __restrict__ shifts, const float* __restrict__ rarr,
                            const float* __restrict__ efb, const float* __restrict__ R1,
                            const float* __restrict__ sU1,
                            float* __restrict__ m0s, float* __restrict__ m10,
                            float* __restrict__ m11, float* __restrict__ m12)
{
    int e = blockIdx.x, k = threadIdx.x;
    __shared__ float sef[NBb]; __shared__ float sY[3]; __shared__ int sIdx[2];
    if (k == 0) {
        int snd = ei[e], rcv = ei[Ee + e];
        sIdx[0] = snd; sIdx[1] = rcv;
        float r = rarr[e];
        float vx = pos[rcv*3+0]-pos[snd*3+0]+shifts[e*3+0];
        float vy = pos[rcv*3+1]-pos[snd*3+1]+shifts[e*3+1];
        float vz = pos[rcv*3+2]-pos[snd*3+2]+shifts[e*3+2];
        sY[0] = SQ3f * vx / r; sY[1] = SQ3f * vy / r; sY[2] = SQ3f * vz / r;
    }
    if (k < NBb) sef[k] = efb[(size_t)e * NBb + k];
    __syncthreads();
    int snd = sIdx[0], rcv = sIdx[1];
    float w0 = 0.f, w1 = 0.f;
#pragma unroll
    for (int b = 0; b < NBb; ++b) {
        float fe = sef[b];
        w0 += fe * R1[b * 256 + k];
        w1 += fe * R1[b * 256 + 128 + k];
    }
    float sj = sU1[(size_t)snd * Kc + k];
    atomicAdd(&m0s[(size_t)rcv * Kc + k], w0 * sj);
    float m1c = w1 * sj;
    atomicAdd(&m10[(size_t)rcv * Kc + k], m1c * sY[0]);
    atomicAdd(&m11[(size_t)rcv * Kc + k], m1c * sY[1]);
    atomicAdd(&m12[(size_t)rcv * Kc + k], m1c * sY[2]);
}

// layer-2 forward edge messages: scatter m0 into s2pre
__global__ void edge_fwd2_k(const int* __restrict__ ei, const float* __restrict__ pos,
                            const float* __restrict__ shifts, const float* __restrict__ rarr,
                            const float* __restrict__ efb, const float* __restrict__ R2,
                            const float* __restrict__ sU2, const float* __restrict__ vu0,
                            const float* __restrict__ vu1, const float* __restrict__ vu2,
                            float* __restrict__ s2pre)
{
    int e = blockIdx.x, k = threadIdx.x;
    __shared__ float sef[NBb]; __shared__ float sY[3]; __shared__ int sIdx[2];
    if (k == 0) {
        int snd = ei[e], rcv = ei[Ee + e];
        sIdx[0] = snd; sIdx[1] = rcv;
        float r = rarr[e];
        float vx = pos[rcv*3+0]-pos[snd*3+0]+shifts[e*3+0];
        float vy = pos[rcv*3+1]-pos[snd*3+1]+shifts[e*3+1];
        float vz = pos[rcv*3+2]-pos[snd*3+2]+shifts[e*3+2];
        sY[0] = SQ3f * vx / r; sY[1] = SQ3f * vy / r; sY[2] = SQ3f * vz / r;
    }
    if (k < NBb) sef[k] = efb[(size_t)e * NBb + k];
    __syncthreads();
    int snd = sIdx[0], rcv = sIdx[1];
    float w00 = 0.f, w110 = 0.f;
#pragma unroll
    for (int b = 0; b < NBb; ++b) {
        float fe = sef[b];
        w00  += fe * R2[b * 512 + k];
        w110 += fe * R2[b * 512 + 128 + k];
    }
    float sj = sU2[(size_t)snd * Kc + k];
    float v0 = vu0[(size_t)snd * Kc + k];
    float v1 = vu1[(size_t)snd * Kc + k];
    float v2 = vu2[(size_t)snd * Kc + k];
    float vdot = v0 * sY[0] + v1 * sY[1] + v2 * sY[2];
    atomicAdd(&s2pre[(size_t)rcv * Kc + k], w00 * sj + w110 * vdot * (1.f / SQ3f));
}

// layer-2 backward edge: scatter g_sj2, g_vj; reduce g_Y1/g_ef -> g_pos
__global__ void edge_bwd2_k(const int* __restrict__ ei, const float* __restrict__ pos,
                            const float* __restrict__ shifts, const float* __restrict__ rarr,
                            const float* __restrict__ efb, const float* __restrict__ R2,
                            const float* __restrict__ sU2, const float* __restrict__ vu0,
                            const float* __restrict__ vu1, const float* __restrict__ vu2,
                            const float* __restrict__ gm0s,
                            float* __restrict__ gup2, float* __restrict__ ga0,
                            float* __restrict__ ga1, float* __restrict__ ga2,
                            float* __restrict__ gpos)
{
    int e = blockIdx.x, k = threadIdx.x;
    int lane = k & 31;
    __shared__ float sef[NBb]; __shared__ float sY[3]; __shared__ int sIdx[2];
    __shared__ float red[11];
    float vx = 0.f, vy = 0.f, vz = 0.f, rr = 1.f;
    if (k == 0) {
        int snd = ei[e], rcv = ei[Ee + e];
        sIdx[0] = snd; sIdx[1] = rcv;
        rr = rarr[e];
        vx = pos[rcv*3+0]-pos[snd*3+0]+shifts[e*3+0];
        vy = pos[rcv*3+1]-pos[snd*3+1]+shifts[e*3+1];
        vz = pos[rcv*3+2]-pos[snd*3+2]+shifts[e*3+2];
        sY[0] = SQ3f * vx / rr; sY[1] = SQ3f * vy / rr; sY[2] = SQ3f * vz / rr;
    }
    if (k < NBb) sef[k] = efb[(size_t)e * NBb + k];
    if (k < 11) red[k] = 0.f;
    __syncthreads();
    int snd = sIdx[0], rcv = sIdx[1];
    float gm0 = gm0s[(size_t)rcv * Kc + k];
    float w00 = 0.f, w110 = 0.f;
#pragma unroll
    for (int b = 0; b < NBb; ++b) {
        float fe = sef[b];
        w00  += fe * R2[b * 512 + k];
        w110 += fe * R2[b * 512 + 128 + k];
    }
    float sj = sU2[(size_t)snd * Kc + k];
    float v0 = vu0[(size_t)snd * Kc + k];
    float v1 = vu1[(size_t)snd * Kc + k];
    float v2 = vu2[(size_t)snd * Kc + k];
    float vdot = v0 * sY[0] + v1 * sY[1] + v2 * sY[2];
    atomicAdd(&gup2[(size_t)snd * Kc + k], gm0 * w00);
    float gvd = gm0 * w110 * (1.f / SQ3f);
    atomicAdd(&ga0[(size_t)snd * Kc + k], gvd * sY[0]);
    atomicAdd(&ga1[(size_t)snd * Kc + k], gvd * sY[1]);
    atomicAdd(&ga2[(size_t)snd * Kc + k], gvd * sY[2]);
    float vals[11];
    vals[0] = gvd * v0; vals[1] = gvd * v1; vals[2] = gvd * v2;
    float gw00 = gm0 * sj, gw110 = gm0 * vdot * (1.f / SQ3f);
#pragma unroll
    for (int b = 0; b < NBb; ++b)
        vals[3 + b] = gw00 * R2[b * 512 + k] + gw110 * R2[b * 512 + 128 + k];
#pragma unroll
    for (int i = 0; i < 11; ++i) {
        float sv = wave_red(vals[i]);
        if (lane == 0) atomicAdd(&red[i], sv);
    }
    __syncthreads();
    if (k == 0) {
        float def[NBb]; ef_deriv(rr, def);
        float g_r = 0.f;
#pragma unroll
        for (int b = 0; b < NBb; ++b) g_r += red[3 + b] * def[b];
        float gY0 = red[0], gY1 = red[1], gY2 = red[2];
        float gdv = gY0 * vx + gY1 * vy + gY2 * vz;
        float ir = 1.f / rr, ir3 = ir * ir * ir;
        float gvx = SQ3f * (gY0 * ir - vx * gdv * ir3) + g_r * vx * ir;
        float gvy = SQ3f * (gY1 * ir - vy * gdv * ir3) + g_r * vy * ir;
        float gvz = SQ3f * (gY2 * ir - vz * gdv * ir3) + g_r * vz * ir;
        atomicAdd(&gpos[rcv*3+0],  gvx); atomicAdd(&gpos[snd*3+0], -gvx);
        atomicAdd(&gpos[rcv*3+1],  gvy); atomicAdd(&gpos[snd*3+1], -gvy);
        atomicAdd(&gpos[rcv*3+2],  gvz); atomicAdd(&gpos[snd*3+2], -gvz);
    }
}

// layer-1 backward edge: reduce g_Y1/g_ef -> g_pos (sj path is position-free)
__global__ void edge_bwd1_k(const int* __restrict__ ei, const float* __restrict__ pos,
                            const float* __restrict__ shifts, const float* __restrict__ rarr,
                            const float* __restrict__ efb, const float* __restrict__ R1,
                            const float* __restrict__ sU1, const float* __restrict__ gm0s,
                            const float* __restrict__ gm10, const float* __restrict__ gm11,
                            const float* __restrict__ gm12, float* __restrict__ gpos)
{
    int e = blockIdx.x, k = threadIdx.x;
    int lane = k & 31;
    __shared__ float sef[NBb]; __shared__ float sY[3]; __shared__ int sIdx[2];
    __shared__ float red[11];
    float vx = 0.f, vy = 0.f, vz = 0.f, rr = 1.f;
    if (k == 0) {
        int snd = ei[e], rcv = ei[Ee + e];
        sIdx[0] = snd; sIdx[1] = rcv;
        rr = rarr[e];
        vx = pos[rcv*3+0]-pos[snd*3+0]+shifts[e*3+0];
        vy = pos[rcv*3+1]-pos[snd*3+1]+shifts[e*3+1];
        vz = pos[rcv*3+2]-pos[snd*3+2]+shifts[e*3+2];
        sY[0] = SQ3f * vx / rr; sY[1] = SQ3f * vy / rr; sY[2] = SQ3f * vz / rr;
    }
    if (k < NBb) sef[k] = efb[(size_t)e * NBb + k];
    if (k < 11) red[k] = 0.f;
    __syncthreads();
    int snd = sIdx[0], rcv = sIdx[1];
    float gm0 = gm0s[(size_t)rcv * Kc + k];
    float g10 = gm10[(size_t)rcv * Kc + k];
    float g11 = gm11[(size_t)rcv * Kc + k];
    float g12 = gm12[(size_t)rcv * Kc + k];
    float w0 = 0.f, w1 = 0.f;
#pragma unroll
    for (int b = 0; b < NBb; ++b) {
        float fe = sef[b];
        w0 += fe * R1[b * 256 + k];
        w1 += fe * R1[b * 256 + 128 + k];
    }
    float sj = sU1[(size_t)snd * Kc + k];
    float gdotc = g10 * sY[0] + g11 * sY[1] + g12 * sY[2];
    float gw0 = gm0 * sj, gw1 = gdotc * sj;
    float w1sj = w1 * sj;
    float vals[11];
    vals[0] = g10 * w1sj; vals[1] = g11 * w1sj; vals[2] = g12 * w1sj;
#pragma unroll
    for (int b = 0; b < NBb; ++b)
        vals[3 + b] = gw0 * R1[b * 256 + k] + gw1 * R1[b * 256 + 128 + k];
#pragma unroll
    for (int i = 0; i < 11; ++i) {
        float sv = wave_red(vals[i]);
        if (lane == 0) atomicAdd(&red[i], sv);
    }
    __syncthreads();
    if (k == 0) {
        float def[NBb]; ef_deriv(rr, def);
        float g_r = 0.f;
#pragma unroll
        for (int b = 0; b < NBb; ++b) g_r += red[3 + b] * def[b];
        float gY0 = red[0], gY1 = red[1], gY2 = red[2];
        float gdv = gY0 * vx + gY1 * vy + gY2 * vz;
        float ir = 1.f / rr, ir3 = ir * ir * ir;
        float gvx = SQ3f * (gY0 * ir - vx * gdv * ir3) + g_r * vx * ir;
        float gvy = SQ3f * (gY1 * ir - vy * gdv * ir3) + g_r * vy * ir;
        float gvz = SQ3f * (gY2 * ir - vz * gdv * ir3) + g_r * vz * ir;
        atomicAdd(&gpos[rcv*3+0],  gvx); atomicAdd(&gpos[snd*3+0], -gvx);
        atomicAdd(&gpos[rcv*3+1],  gvy); atomicAdd(&gpos[snd*3+1], -gvy);
        atomicAdd(&gpos[rcv*3+2],  gvz); atomicAdd(&gpos[snd*3+2], -gvz);
    }
}

// e1 = seg(s1 @ w_r1, batch);  one wave per node
__global__ void e1_readout_k(const float* __restrict__ s1, const float* __restrict__ w_r1,
                             const int* __restrict__ batch, float* __restrict__ e1g)
{
    int lane = threadIdx.x & 31, wave = threadIdx.x >> 5;
    int node = blockIdx.x * 8 + wave;
    if (node >= Nn) return;
    float acc = 0.f;
    for (int k = lane; k < Kc; k += 32) acc += s1[(size_t)node * Kc + k] * w_r1[k];
    acc = wave_red(acc);
    if (lane == 0) atomicAdd(&e1g[batch[node]], acc);
}

// e2 readout + gcoef[n][j] = w_out[j]*silu'(h[n][j]);  one wave per node
__global__ void readout2_k(const float* __restrict__ s2, const float* __restrict__ W_mlp,
                           const float* __restrict__ w_out, const int* __restrict__ batch,
                           float* __restrict__ e2g, float* __restrict__ gcoef)
{
    int lane = threadIdx.x & 31, wave = threadIdx.x >> 5;
    int node = blockIdx.x * 8 + wave;
    if (node >= Nn) return;
    int j = lane & 15, half = lane >> 4;
    float h = 0.f;
    for (int t = 0; t < 64; ++t) {
        int k = half * 64 + t;
        h += s2[(size_t)node * Kc + k] * W_mlp[k * MLPDc + j];
    }
    h += __shfl_xor(h, 16, 32);                       // both halves hold full h_j
    float sig = 1.f / (1.f + __expf(-h));
    float a = h * sig;
    float wo = w_out[j];
    float p = a * wo;
    p += __shfl_xor(p, 8, 32); p += __shfl_xor(p, 4, 32);
    p += __shfl_xor(p, 2, 32); p += __shfl_xor(p, 1, 32);
    if (lane == 0) atomicAdd(&e2g[batch[node]], p);
    if (lane < 16) gcoef[(size_t)node * MLPDc + j] = wo * sig * (1.f + h * (1.f - sig));
}

__global__ void e0_k(const int* __restrict__ z, const int* __restrict__ batch,
                     const float* __restrict__ ae, float* __restrict__ e0g)
{
    int n = blockIdx.x * blockDim.x + threadIdx.x;
    if (n >= Nn) return;
    atomicAdd(&e0g[batch[n]], ae[z[n]]);
}

__global__ void gs1_init_k(const float* __restrict__ w_r1, float* __restrict__ gs1)
{
    int i = blockIdx.x * blockDim.x + threadIdx.x;
    if (i >= Nn * Kc) return;
    gs1[i] = w_r1[i & (Kc - 1)];
}

__global__ void finalize_scalars_k(const float* __restrict__ eg, float* __restrict__ out)
{
    int t = threadIdx.x;
    if (t < Gg) {
        float a = eg[t], b = eg[Gg + t], c = eg[2 * Gg + t];
        out[t] = a + b + c;
        out[Gg + t * 3 + 0] = a;
        out[Gg + t * 3 + 1] = b;
        out[Gg + t * 3 + 2] = c;
    }
}

__global__ void finalize_forces_k(const float* __restrict__ gpos, float* __restrict__ out)
{
    int i = blockIdx.x * blockDim.x + threadIdx.x;
    if (i < Nn * 3) out[Gg + Gg * 3 + i] = -gpos[i];
}

// ---------------------------------------------------------------------------
extern "C" void kernel_launch(void* const* d_in, const int* in_sizes, int n_in,
                              void* d_out, int out_size, void* d_ws, size_t ws_size,
                              hipStream_t stream)
{
    (void)in_sizes; (void)n_in; (void)out_size; (void)ws_size;
    const float* positions = (const float*)d_in[0];
    const float* shifts    = (const float*)d_in[1];
    const int*   ei        = (const int*)d_in[2];
    const int*   node_z    = (const int*)d_in[3];
    const int*   batch     = (const int*)d_in[4];
    const float* atomic_e  = (const float*)d_in[5];
    const float* W_embed   = (const float*)d_in[6];
    const float* R1        = (const float*)d_in[7];
    const float* W_up1     = (const float*)d_in[8];
    const float* W_mix_s1  = (const float*)d_in[9];
    const float* W_mix_v1  = (const float*)d_in[10];
    const float* W_sk_s1   = (const float*)d_in[11];
    const float* w_r1      = (const float*)d_in[12];
    const float* R2        = (const float*)d_in[13];
    const float* W_up2s    = (const float*)d_in[14];
    const float* W_up2v    = (const float*)d_in[15];
    const float* W_mix_s2  = (const float*)d_in[16];
    const float* W_sk_s2   = (const float*)d_in[17];
    const float* W_mlp     = (const float*)d_in[18];
    const float* w_out     = (const float*)d_in[19];
    float* out = (float*)d_out;

    const size_t PL = (size_t)Nn * Kc;
    float* ws = (float*)d_ws;
    auto P = [&](int i) { return ws + (size_t)i * PL; };
    size_t off = 17 * PL;
    float* efb  = ws + off; off += (size_t)Ee * NBb;
    float* rb   = ws + off; off += Ee;
    float* gco  = ws + off; off += (size_t)Nn * MLPDc;
    float* gpos = ws + off; off += (size_t)Nn * 3;
    float* eg   = ws + off; off += 3 * Gg;          // e0g | e1g | e2g

    const int ZT = 256;
    auto zg = [&](size_t n) { return dim3((unsigned)((n + ZT - 1) / ZT)); };
    dim3 gB(Nn / 16), tB(256);                      // 625 row tiles x 8 waves
    dim3 gE(Ee), tE(128);
    dim3 gW((Nn + 7) / 8);                          // wave-per-node readouts

    // ---- zero accumulation targets (planes 2..5 = m0sum,m1sum; gpos+eg) ----
    zero_k<<<zg(4 * PL), ZT, 0, stream>>>(P(2), 4 * PL);
    zero_k<<<zg((size_t)Nn * 3 + 3 * Gg), ZT, 0, stream>>>(gpos, (size_t)Nn * 3 + 3 * Gg);

    // ---- forward ----
    embed_k<<<zg(PL), ZT, 0, stream>>>(node_z, W_embed, P(0));                        // s
    gemm_wmma_t<Kc,Kc,0><<<gB, tB, 0, stream>>>(P(0), W_up1, P(1), 1.f, 0);           // sU1
    edge_geom_k<<<zg(Ee), ZT, 0, stream>>>(ei, positions, shifts, rb, efb);
    edge_fwd1_k<<<gE, tE, 0, stream>>>(ei, positions, shifts, rb, efb, R1, P(1),
                                       P(2), P(3), P(4), P(5));
    gemm_wmma_t<Kc,Kc,0><<<gB, tB, 0, stream>>>(P(2), W_mix_s1, P(6), 1.f/AVGf, 0);   // s1
    gemm_sk_t<0><<<gB, tB, 0, stream>>>(P(0), W_sk_s1, node_z, P(6));                 // s1 += skip
    for (int x = 0; x < 3; ++x)                                                       // v1
        gemm_wmma_t<Kc,Kc,0><<<gB, tB, 0, stream>>>(P(3+x), W_mix_v1, P(7+x), 1.f/AVGf, 0);
    e1_readout_k<<<gW, tB, 0, stream>>>(P(6), w_r1, batch, eg + Gg);
    gemm_wmma_t<Kc,Kc,0><<<gB, tB, 0, stream>>>(P(6), W_up2s, P(10), 1.f, 0);         // sU2
    for (int x = 0; x < 3; ++x)                                                       // vup2
        gemm_wmma_t<Kc,Kc,0><<<gB, tB, 0, stream>>>(P(7+x), W_up2v, P(11+x), 1.f, 0);
    zero_k<<<zg(PL), ZT, 0, stream>>>(P(2), PL);                                      // s2pre (reuse)
    edge_fwd2_k<<<gE, tE, 0, stream>>>(ei, positions, shifts, rb, efb, R2,
                                       P(10), P(11), P(12), P(13), P(2));
    gemm_wmma_t<Kc,Kc,0><<<gB, tB, 0, stream>>>(P(2), W_mix_s2, P(0), 1.f/AVGf, 0);   // s2 (reuse P0)
    gemm_sk_t<0><<<gB, tB, 0, stream>>>(P(6), W_sk_s2, node_z, P(0));                 // s2 += skip
    readout2_k<<<gW, tB, 0, stream>>>(P(0), W_mlp, w_out, batch, eg + 2 * Gg, gco);

    // ---- backward ----
    // g_s2 = gcoef @ W_mlp^T   (kd = 16)
    gemm_wmma_t<MLPDc,MLPDc,1><<<gB, tB, 0, stream>>>(gco, W_mlp, P(14), 1.f, 0);
    // g_s1 = w_r1 (broadcast)  + g_s2 @ W_sk_s2[z]^T
    gs1_init_k<<<zg(PL), ZT, 0, stream>>>(w_r1, P(15));
    gemm_sk_t<1><<<gB, tB, 0, stream>>>(P(14), W_sk_s2, node_z, P(15));
    // g_m0sum2 = g_s2 @ W_mix_s2^T / AVG
    gemm_wmma_t<Kc,Kc,1><<<gB, tB, 0, stream>>>(P(14), W_mix_s2, P(2), 1.f/AVGf, 0);
    // zero edge-scatter targets for bwd2
    zero_k<<<zg(3 * PL), ZT, 0, stream>>>(P(3), 3 * PL);
    zero_k<<<zg(PL), ZT, 0, stream>>>(P(16), PL);
    edge_bwd2_k<<<gE, tE, 0, stream>>>(ei, positions, shifts, rb, efb, R2,
                                       P(10), P(11), P(12), P(13), P(2),
                                       P(16), P(3), P(4), P(5), gpos);
    // g_s1 += g_up2acc @ W_up2s^T
    gemm_wmma_t<Kc,Kc,1><<<gB, tB, 0, stream>>>(P(16), W_up2s, P(15), 1.f, 1);
    // g_v1 = g_vupacc @ W_up2v^T ; g_m1sum = g_v1 @ W_mix_v1^T / AVG
    for (int x = 0; x < 3; ++x)
        gemm_wmma_t<Kc,Kc,1><<<gB, tB, 0, stream>>>(P(3+x), W_up2v, P(7+x), 1.f, 0);
    for (int x = 0; x < 3; ++x)
        gemm_wmma_t<Kc,Kc,1><<<gB, tB, 0, stream>>>(P(7+x), W_mix_v1, P(11+x), 1.f/AVGf, 0);
    // g_m0sum1 = g_s1 @ W_mix_s1^T / AVG
    gemm_wmma_t<Kc,Kc,1><<<gB, tB, 0, stream>>>(P(15), W_mix_s1, P(10), 1.f/AVGf, 0);
    edge_bwd1_k<<<gE, tE, 0, stream>>>(ei, positions, shifts, rb, efb, R1,
                                       P(1), P(10), P(11), P(12), P(13), gpos);

    // ---- readout / outputs ----
    e0_k<<<zg(Nn), ZT, 0, stream>>>(node_z, batch, atomic_e, eg);
    finalize_scalars_k<<<dim3(1), dim3(64), 0, stream>>>(eg, out);
    finalize_forces_k<<<zg((size_t)Nn * 3), ZT, 0, stream>>>(gpos, out);
}